// vPFAE_Transformer_16209206575872
// MI455X (gfx1250) — compile-verified
//
#include <hip/hip_runtime.h>
#include <math.h>

typedef float v2f __attribute__((ext_vector_type(2)));
typedef float v8f __attribute__((ext_vector_type(8)));

// ---------------------------------------------------------------------------
// atomic float max via int/uint ordering trick (safe on all ROCm versions)
// ---------------------------------------------------------------------------
__device__ __forceinline__ void atomicMaxF(float* addr, float val) {
  if (val >= 0.0f)
    atomicMax((int*)addr, __float_as_int(val));
  else
    atomicMin((unsigned int*)addr, (unsigned int)__float_as_int(val));
}

// ---------------------------------------------------------------------------
// WMMA f32 GEMM:  C[M x Ncols] = A[M x K] @ W[K x Ncols] (+bias)
// one wave32 per 16x16 tile, K stepped by 4 via V_WMMA_F32_16X16X4_F32.
// Requires M % 16 == 0, Ncols % 16 == 0, K % 4 == 0 (true for all calls).
// A fragment layout (16x4 f32): lanes 0-15 hold M=0..15; VGPR0 = K+2*half,
// VGPR1 = K+2*half+1 where half = lane>>4. B mirrors with N across lanes.
// C/D: VGPR r -> row r + 8*half, col = lane&15.
// ---------------------------------------------------------------------------
__global__ __launch_bounds__(256) void wmma_gemm_f32(
    const float* __restrict__ A, const float* __restrict__ W,
    const float* __restrict__ bias, float* __restrict__ C,
    int M, int K, int Ncols, int ldc, int col_off) {
  int wave = (int)((blockIdx.x * blockDim.x + threadIdx.x) >> 5);
  int lane = threadIdx.x & 31;
  int tilesN = Ncols >> 4;
  int tm = wave / tilesN;
  int tn = wave - tm * tilesN;
  int row0 = tm << 4;
  if (row0 >= M) return;
  int col0 = tn << 4;
  int half = lane >> 4;
  int l = lane & 15;

  const float* arow = A + (size_t)(row0 + l) * K;
  const float* wcol = W + col0 + l;
  v8f acc = {};
  for (int k = 0; k < K; k += 4) {
    v2f a = *(const v2f*)(arow + k + 2 * half);   // 8B aligned (k, half even mults)
    v2f b;
    b.x = wcol[(size_t)(k + 2 * half) * Ncols];
    b.y = wcol[(size_t)(k + 2 * half + 1) * Ncols];
    acc = __builtin_amdgcn_wmma_f32_16x16x4_f32(
        false, a, false, b, (short)0, acc, false, false);
  }
  float bv = bias ? bias[col0 + l] : 0.0f;
  float* crow = C + (size_t)(row0 + 8 * half) * ldc + col_off + col0 + l;
#pragma unroll
  for (int r = 0; r < 8; ++r) crow[(size_t)r * ldc] = acc[r] + bv;
}

// ---------------------------------------------------------------------------
// elementwise helpers
// ---------------------------------------------------------------------------
__global__ void fill_kernel(float* __restrict__ p, float v, int n) {
  int i = blockIdx.x * blockDim.x + threadIdx.x;
  if (i < n) p[i] = v;
}
__global__ void relu_inplace(float* __restrict__ p, int n) {
  int i = blockIdx.x * blockDim.x + threadIdx.x;
  if (i < n) p[i] = fmaxf(p[i], 0.0f);
}
__global__ void rsqrt_inplace(float* __restrict__ p, int n) {
  int i = blockIdx.x * blockDim.x + threadIdx.x;
  if (i < n) p[i] = rsqrtf(p[i]);
}

// ---------------------------------------------------------------------------
// Pass 1: per-edge score = <q[dst], k[src] + ea@We> / sqrt(C); atomic segment max
// One wave32 per edge; We (16 x C) staged in LDS; channels strided by 32.
// ---------------------------------------------------------------------------
__global__ __launch_bounds__(256) void edge_score(
    const int* __restrict__ src, const int* __restrict__ dst,
    const float* __restrict__ ea, const float* __restrict__ We,
    const float* __restrict__ q, const float* __restrict__ k,
    float* __restrict__ score, float* __restrict__ mbuf,
    int E, int C, float scale) {
  __shared__ float sWe[16 * 96];
  for (int i = threadIdx.x; i < 16 * C; i += blockDim.x) sWe[i] = We[i];
  __syncthreads();
  int e = (int)((blockIdx.x * blockDim.x + threadIdx.x) >> 5);
  int lane = threadIdx.x & 31;
  if (e >= E) return;
  int s = src[e], d = dst[e];
  float eav[16];
  const float* ep = ea + (size_t)e * 16;
#pragma unroll
  for (int j = 0; j < 16; ++j) eav[j] = ep[j];   // uniform across wave -> scalarized
  const float* kp = k + (size_t)s * C;
  const float* qp = q + (size_t)d * C;
  float part = 0.0f;
  for (int c = lane; c < C; c += 32) {
    float ec = 0.0f;
#pragma unroll
    for (int j = 0; j < 16; ++j) ec = fmaf(eav[j], sWe[j * C + c], ec);
    part = fmaf(qp[c], kp[c] + ec, part);
  }
#pragma unroll
  for (int off = 16; off > 0; off >>= 1) part += __shfl_xor(part, off, 32);
  if (lane == 0) {
    float sc = part * scale;
    score[e] = sc;
    atomicMaxF(&mbuf[d], sc);
  }
}

// ---------------------------------------------------------------------------
// Pass 2: ex = exp(score - m[dst]); denom[dst] += ex   (score overwritten)
// ---------------------------------------------------------------------------
__global__ void edge_expsum(const int* __restrict__ dst,
                            const float* __restrict__ mbuf,
                            float* __restrict__ score,
                            float* __restrict__ denom, int E) {
  int e = blockIdx.x * blockDim.x + threadIdx.x;
  if (e >= E) return;
  int d = dst[e];
  float ex = expf(score[e] - mbuf[d]);
  score[e] = ex;
  atomicAdd(&denom[d], ex);
}

// ---------------------------------------------------------------------------
// Pass 3: out[dst] += alpha * (v[src] + ea@We);  out pre-loaded with skip.
// ---------------------------------------------------------------------------
__global__ __launch_bounds__(256) void edge_aggregate(
    const int* __restrict__ src, const int* __restrict__ dst,
    const float* __restrict__ ea, const float* __restrict__ We,
    const float* __restrict__ v, const float* __restrict__ ex,
    const float* __restrict__ denom, float* __restrict__ out,
    int E, int C) {
  __shared__ float sWe[16 * 96];
  for (int i = threadIdx.x; i < 16 * C; i += blockDim.x) sWe[i] = We[i];
  __syncthreads();
  int e = (int)((blockIdx.x * blockDim.x + threadIdx.x) >> 5);
  int lane = threadIdx.x & 31;
  if (e >= E) return;
  int s = src[e], d = dst[e];
  float alpha = ex[e] / (denom[d] + 1e-16f);
  float eav[16];
  const float* ep = ea + (size_t)e * 16;
#pragma unroll
  for (int j = 0; j < 16; ++j) eav[j] = ep[j];
  const float* vp = v + (size_t)s * C;
  float* op = out + (size_t)d * C;
  for (int c = lane; c < C; c += 32) {
    float ec = 0.0f;
#pragma unroll
    for (int j = 0; j < 16; ++j) ec = fmaf(eav[j], sWe[j * C + c], ec);
    atomicAdd(op + c, alpha * (vp[c] + ec));
  }
}

// ---------------------------------------------------------------------------
// GCN pieces
// ---------------------------------------------------------------------------
__global__ void deg_kernel(const int* __restrict__ dst, float* __restrict__ degf,
                           int E) {
  int e = blockIdx.x * blockDim.x + threadIdx.x;
  if (e < E) atomicAdd(&degf[dst[e]], 1.0f);
}

// agg[i,c] = dinv[i]^2 * g[i,c]   (self-loop term), g/agg are [N x 64]
__global__ void gcn_selfinit(const float* __restrict__ g,
                             const float* __restrict__ dinv,
                             float* __restrict__ agg, int total) {
  int i = blockIdx.x * blockDim.x + threadIdx.x;
  if (i >= total) return;
  float di = dinv[i >> 6];
  agg[i] = di * di * g[i];
}

// one wave per edge, 64 channels: agg[dst] += dinv[src]*dinv[dst]*g[src]
__global__ __launch_bounds__(256) void gcn_scatter(
    const int* __restrict__ src, const int* __restrict__ dst,
    const float* __restrict__ dinv, const float* __restrict__ g,
    float* __restrict__ agg, int E) {
  int e = (int)((blockIdx.x * blockDim.x + threadIdx.x) >> 5);
  int lane = threadIdx.x & 31;
  if (e >= E) return;
  int s = src[e], d = dst[e];
  float w = dinv[s] * dinv[d];
  const float* gp = g + (size_t)s * 64;
  float* ap = agg + (size_t)d * 64;
  atomicAdd(ap + lane, w * gp[lane]);
  atomicAdd(ap + lane + 32, w * gp[lane + 32]);
}

// d_out = [mu (N x 32) | logstd (N x 32)], agg is [N x 64] = [mu|ls]
__global__ void gcn_finalize(const float* __restrict__ agg,
                             const float* __restrict__ bmu,
                             const float* __restrict__ bls,
                             float* __restrict__ out, int N) {
  int i = blockIdx.x * blockDim.x + threadIdx.x;
  if (i >= N * 32) return;
  int n = i >> 5, c = i & 31;
  out[i] = agg[(size_t)n * 64 + c] + bmu[c];
  out[(size_t)N * 32 + i] = agg[(size_t)n * 64 + 32 + c] + bls[c];
}

// ---------------------------------------------------------------------------
// host launcher
// ---------------------------------------------------------------------------
static inline int cdiv(int a, int b) { return (a + b - 1) / b; }

extern "C" void kernel_launch(void* const* d_in, const int* in_sizes, int n_in,
                              void* d_out, int out_size, void* d_ws, size_t ws_size,
                              hipStream_t stream) {
  const float* x   = (const float*)d_in[0];
  const int*   ei  = (const int*)d_in[1];
  const float* ea  = (const float*)d_in[2];
  const float* Wq1 = (const float*)d_in[3];  const float* bq1 = (const float*)d_in[4];
  const float* Wk1 = (const float*)d_in[5];  const float* bk1 = (const float*)d_in[6];
  const float* Wv1 = (const float*)d_in[7];  const float* bv1 = (const float*)d_in[8];
  const float* We1 = (const float*)d_in[9];
  const float* Ws1 = (const float*)d_in[10]; const float* bs1 = (const float*)d_in[11];
  const float* Wq2 = (const float*)d_in[12]; const float* bq2 = (const float*)d_in[13];
  const float* Wk2 = (const float*)d_in[14]; const float* bk2 = (const float*)d_in[15];
  const float* Wv2 = (const float*)d_in[16]; const float* bv2 = (const float*)d_in[17];
  const float* We2 = (const float*)d_in[18];
  const float* Ws2 = (const float*)d_in[19]; const float* bs2 = (const float*)d_in[20];
  const float* Wmu = (const float*)d_in[21]; const float* bmu = (const float*)d_in[22];
  const float* Wls = (const float*)d_in[23]; const float* bls = (const float*)d_in[24];

  const int N = in_sizes[0] / 128;   // 50000
  const int E = in_sizes[2] / 16;    // 1600000
  const int* srcI = ei;
  const int* dstI = ei + E;

  // workspace carve-up (floats)
  float* ws    = (float*)d_ws;
  float* qb    = ws;                       // N*96
  float* kb    = qb  + (size_t)N * 96;     // N*96
  float* vb    = kb  + (size_t)N * 96;     // N*96
  float* acc1  = vb  + (size_t)N * 96;     // N*96  skip accumulator L1 -> h1
  float* acc2  = acc1 + (size_t)N * 96;    // N*96  skip accumulator L2 -> h2
  float* score = acc2 + (size_t)N * 96;    // E
  float* mb    = score + (size_t)E;        // N (seg-max, later deg/dinv)
  float* dn    = mb + (size_t)N;           // N (seg-denom)

  const int TB = 256;
  const int edgeWaveBlocks = cdiv(E, TB / 32);
  const int edgeBlocks = cdiv(E, TB);

  auto gemm = [&](const float* A, const float* W, const float* bias, float* C,
                  int K, int Ncols, int ldc, int coff) {
    int tiles = (N / 16) * (Ncols / 16);
    wmma_gemm_f32<<<cdiv(tiles * 32, TB), TB, 0, stream>>>(A, W, bias, C, N, K,
                                                           Ncols, ldc, coff);
  };

  // ---------------- TransformerConv layer 1: 128 -> 96 ----------------
  {
    const int C = 96;
    gemm(x, Wq1, bq1, qb,   128, C, C, 0);
    gemm(x, Wk1, bk1, kb,   128, C, C, 0);
    gemm(x, Wv1, bv1, vb,   128, C, C, 0);
    gemm(x, Ws1, bs1, acc1, 128, C, C, 0);   // skip -> accumulator init
    fill_kernel<<<cdiv(N, TB), TB, 0, stream>>>(mb, -INFINITY, N);
    fill_kernel<<<cdiv(N, TB), TB, 0, stream>>>(dn, 0.0f, N);
    edge_score<<<edgeWaveBlocks, TB, 0, stream>>>(srcI, dstI, ea, We1, qb, kb,
                                                  score, mb, E, C,
                                                  rsqrtf((float)C));
    edge_expsum<<<edgeBlocks, TB, 0, stream>>>(dstI, mb, score, dn, E);
    edge_aggregate<<<edgeWaveBlocks, TB, 0, stream>>>(srcI, dstI, ea, We1, vb,
                                                      score, dn, acc1, E, C);
    relu_inplace<<<cdiv(N * C, TB), TB, 0, stream>>>(acc1, N * C);  // acc1 = h1
  }

  // ---------------- TransformerConv layer 2: 96 -> 64 ----------------
  {
    const int C = 64;
    gemm(acc1, Wq2, bq2, qb,   96, C, C, 0);
    gemm(acc1, Wk2, bk2, kb,   96, C, C, 0);
    gemm(acc1, Wv2, bv2, vb,   96, C, C, 0);
    gemm(acc1, Ws2, bs2, acc2, 96, C, C, 0);
    fill_kernel<<<cdiv(N, TB), TB, 0, stream>>>(mb, -INFINITY, N);
    fill_kernel<<<cdiv(N, TB), TB, 0, stream>>>(dn, 0.0f, N);
    edge_score<<<edgeWaveBlocks, TB, 0, stream>>>(srcI, dstI, ea, We2, qb, kb,
                                                  score, mb, E, C,
                                                  rsqrtf((float)C));
    edge_expsum<<<edgeBlocks, TB, 0, stream>>>(dstI, mb, score, dn, E);
    edge_aggregate<<<edgeWaveBlocks, TB, 0, stream>>>(srcI, dstI, ea, We2, vb,
                                                      score, dn, acc2, E, C);
    relu_inplace<<<cdiv(N * C, TB), TB, 0, stream>>>(acc2, N * C);  // acc2 = h2
  }

  // ---------------- GCNConv (mu | logstd): 64 -> 32+32 ----------------
  {
    // degree (in-degree + self loop) -> dinv, stored in mb
    fill_kernel<<<cdiv(N, TB), TB, 0, stream>>>(mb, 1.0f, N);
    deg_kernel<<<edgeBlocks, TB, 0, stream>>>(dstI, mb, E);
    rsqrt_inplace<<<cdiv(N, TB), TB, 0, stream>>>(mb, N);
    // g = h2 @ [Wmu | Wls]  -> qb as [N x 64]
    gemm(acc2, Wmu, nullptr, qb, 64, 32, 64, 0);
    gemm(acc2, Wls, nullptr, qb, 64, 32, 64, 32);
    // agg (kb) = self-loop term + edge scatter
    gcn_selfinit<<<cdiv(N * 64, TB), TB, 0, stream>>>(qb, mb, kb, N * 64);
    gcn_scatter<<<edgeWaveBlocks, TB, 0, stream>>>(srcI, dstI, mb, qb, kb, E);
    gcn_finalize<<<cdiv(N * 32, TB), TB, 0, stream>>>(kb, bmu, bls,
                                                      (float*)d_out, N);
  }
}